// GAT_13649406066804
// MI455X (gfx1250) — compile-verified
//
#include <hip/hip_runtime.h>
#include <math.h>

typedef __attribute__((ext_vector_type(2))) float v2f;
typedef __attribute__((ext_vector_type(8))) float v8f;

#define NEG_SLOPE 0.2f

// ---------------- WMMA wrapper: D = A(16x4) x B(4x16) + C, fp32 ----------------
__device__ __forceinline__ v8f wmma4(v2f a, v2f b, v8f c) {
  // 8 args: (neg_a, A, neg_b, B, c_mod, C, reuse_a, reuse_b)
  return __builtin_amdgcn_wmma_f32_16x16x4_f32(false, a, false, b, (short)0, c,
                                               false, false);
}

// ------------- ordered-uint mapping for float atomicMax (identity = 0) --------
__device__ __forceinline__ unsigned fkey(float f) {
  unsigned b = __float_as_uint(f);
  return (b & 0x80000000u) ? ~b : (b | 0x80000000u);
}
__device__ __forceinline__ float funkey(unsigned k) {
  unsigned b = (k & 0x80000000u) ? (k & 0x7FFFFFFFu) : ~k;
  return __uint_as_float(b);
}
__device__ __forceinline__ float lrelu(float x) {
  return x > 0.f ? x : NEG_SLOPE * x;
}

// ---------------- GEMM1: h1[N,64] = (x .* seq)[N,128] @ W1[128,64] ------------
__global__ void k_gemm1(const float* __restrict__ x, const float* __restrict__ seq,
                        const float* __restrict__ W1, float* __restrict__ h1, int N) {
  __shared__ float lw[64 * 130];  // W1 transposed: lw[col*130 + k], pad keeps b64 aligned
  for (int i = threadIdx.x; i < 128 * 64; i += 256) {
    int k = i >> 6, c = i & 63;
    lw[c * 130 + k] = W1[i];
  }
  __syncthreads();
  const int lane = threadIdx.x & 31;
  const int wave = threadIdx.x >> 5;
  const int l15  = lane & 15;
  const int half = lane >> 4;
  const int row0 = (blockIdx.x * 8 + wave) * 16;
  if (row0 >= N) return;                       // wave-uniform exit (EXEC stays all-1)
  const int arow = row0 + l15;
  const float sq = seq[arow];
  const float* xr = x + (size_t)arow * 128;
  v8f c0 = {}, c1 = {}, c2 = {}, c3 = {};
  for (int k = 0; k < 128; k += 4) {
    const int kk = k + 2 * half;
    v2f a = *(const v2f*)(xr + kk);
    a *= sq;
    v2f b0 = *(const v2f*)(lw + (0  + l15) * 130 + kk);
    v2f b1 = *(const v2f*)(lw + (16 + l15) * 130 + kk);
    v2f b2 = *(const v2f*)(lw + (32 + l15) * 130 + kk);
    v2f b3 = *(const v2f*)(lw + (48 + l15) * 130 + kk);
    c0 = wmma4(a, b0, c0);
    c1 = wmma4(a, b1, c1);
    c2 = wmma4(a, b2, c2);
    c3 = wmma4(a, b3, c3);
  }
  for (int v = 0; v < 8; ++v) {
    int node = row0 + 8 * half + v;
    float* o = h1 + (size_t)node * 64 + l15;
    o[0] = c0[v]; o[16] = c1[v]; o[32] = c2[v]; o[48] = c3[v];
  }
}

// ------------- per-node attention coefficients, layer 1 (8 heads x 8 ch) ------
__global__ void k_att1(const float* __restrict__ h1, const float* __restrict__ asw,
                       const float* __restrict__ adw, float* __restrict__ a_src,
                       float* __restrict__ a_dst, int N) {
  int n = blockIdx.x * blockDim.x + threadIdx.x;
  if (n >= N) return;
  const float* hp = h1 + (size_t)n * 64;
  for (int h = 0; h < 8; ++h) {
    float s = 0.f, d = 0.f;
    for (int c = 0; c < 8; ++c) {
      float v = hp[h * 8 + c];
      s += v * asw[h * 8 + c];
      d += v * adw[h * 8 + c];
    }
    a_src[n * 8 + h] = s;
    a_dst[n * 8 + h] = d;
  }
}

// ----------------------------- edge passes, layer 1 ---------------------------
__device__ __forceinline__ void edge_sd(const int* __restrict__ edges, int e, int E,
                                        int& s, int& d) {
  if (e < E) { s = edges[e]; d = edges[E + e]; }
  else       { s = e - E;    d = s; }          // appended self-loops
}

__global__ void k_edge1_max(const int* __restrict__ edges, const float* __restrict__ a_src,
                            const float* __restrict__ a_dst, unsigned* __restrict__ m1,
                            int E, int N) {
  int e = blockIdx.x * blockDim.x + threadIdx.x;
  if (e >= E + N) return;
  int s, d; edge_sd(edges, e, E, s, d);
  for (int h = 0; h < 8; ++h) {
    float lg = lrelu(a_src[s * 8 + h] + a_dst[d * 8 + h]);
    atomicMax(&m1[d * 8 + h], fkey(lg));
  }
}

__global__ void k_edge1_sum(const int* __restrict__ edges, const float* __restrict__ a_src,
                            const float* __restrict__ a_dst, const unsigned* __restrict__ m1,
                            float* __restrict__ s1, int E, int N) {
  int e = blockIdx.x * blockDim.x + threadIdx.x;
  if (e >= E + N) return;
  int s, d; edge_sd(edges, e, E, s, d);
  for (int h = 0; h < 8; ++h) {
    float lg = lrelu(a_src[s * 8 + h] + a_dst[d * 8 + h]);
    atomicAdd(&s1[d * 8 + h], __expf(lg - funkey(m1[d * 8 + h])));
  }
}

// one wave per edge; 2 features per lane
__global__ void k_edge1_scatter(const int* __restrict__ edges, const float* __restrict__ a_src,
                                const float* __restrict__ a_dst, const unsigned* __restrict__ m1,
                                const float* __restrict__ s1, const float* __restrict__ h1,
                                float* __restrict__ out1, int E, int N) {
  int e = blockIdx.x * 8 + (threadIdx.x >> 5);
  if (e >= E + N) return;
  int lane = threadIdx.x & 31;
  int s, d; edge_sd(edges, e, E, s, d);
  #pragma unroll
  for (int f = lane; f < 64; f += 32) {
    int h = f >> 3;
    float lg    = lrelu(a_src[s * 8 + h] + a_dst[d * 8 + h]);
    float alpha = __expf(lg - funkey(m1[d * 8 + h])) / (s1[d * 8 + h] + 1e-16f);
    atomicAdd(out1 + (size_t)d * 64 + f, alpha * h1[(size_t)s * 64 + f]);
  }
}

// ------------------------------ bias + ELU (in place) -------------------------
__global__ void k_elu(float* __restrict__ o, const float* __restrict__ b1, int total) {
  int i = blockIdx.x * blockDim.x + threadIdx.x;
  if (i >= total) return;
  float v = o[i] + b1[i & 63];
  o[i] = v > 0.f ? v : (__expf(v) - 1.f);
}

// ---------------- GEMM2: h2[N,128] = h[N,64] @ W2[64,128] ---------------------
__global__ void k_gemm2(const float* __restrict__ hin, const float* __restrict__ W2,
                        float* __restrict__ h2, int N) {
  __shared__ float lw[128 * 66];  // W2 transposed: lw[col*66 + k]
  for (int i = threadIdx.x; i < 64 * 128; i += 256) {
    int k = i >> 7, c = i & 127;
    lw[c * 66 + k] = W2[i];
  }
  __syncthreads();
  const int lane = threadIdx.x & 31;
  const int wave = threadIdx.x >> 5;
  const int l15  = lane & 15;
  const int half = lane >> 4;
  const int row0 = (blockIdx.x * 8 + wave) * 16;
  if (row0 >= N) return;
  const float* ar = hin + (size_t)(row0 + l15) * 64;
  v8f acc[8] = {};
  for (int k = 0; k < 64; k += 4) {
    const int kk = k + 2 * half;
    v2f a = *(const v2f*)(ar + kk);
    #pragma unroll
    for (int t = 0; t < 8; ++t) {
      v2f b = *(const v2f*)(lw + (t * 16 + l15) * 66 + kk);
      acc[t] = wmma4(a, b, acc[t]);
    }
  }
  for (int v = 0; v < 8; ++v) {
    int node = row0 + 8 * half + v;
    float* o = h2 + (size_t)node * 128 + l15;
    #pragma unroll
    for (int t = 0; t < 8; ++t) o[t * 16] = acc[t][v];
  }
}

// -------------- per-node attention coefficients, layer 2 (1 head) -------------
__global__ void k_att2(const float* __restrict__ h2, const float* __restrict__ asw,
                       const float* __restrict__ adw, float* __restrict__ a_src,
                       float* __restrict__ a_dst, int N) {
  int n = blockIdx.x * blockDim.x + threadIdx.x;
  if (n >= N) return;
  const float* hp = h2 + (size_t)n * 128;
  float s = 0.f, d = 0.f;
  for (int c = 0; c < 128; ++c) {
    float v = hp[c];
    s += v * asw[c];
    d += v * adw[c];
  }
  a_src[n] = s;
  a_dst[n] = d;
}

// ----------------------------- edge passes, layer 2 ---------------------------
__global__ void k_edge2_max(const int* __restrict__ edges, const float* __restrict__ a_src,
                            const float* __restrict__ a_dst, unsigned* __restrict__ m2,
                            int E, int N) {
  int e = blockIdx.x * blockDim.x + threadIdx.x;
  if (e >= E + N) return;
  int s, d; edge_sd(edges, e, E, s, d);
  atomicMax(&m2[d], fkey(lrelu(a_src[s] + a_dst[d])));
}

__global__ void k_edge2_sum(const int* __restrict__ edges, const float* __restrict__ a_src,
                            const float* __restrict__ a_dst, const unsigned* __restrict__ m2,
                            float* __restrict__ s2, int E, int N) {
  int e = blockIdx.x * blockDim.x + threadIdx.x;
  if (e >= E + N) return;
  int s, d; edge_sd(edges, e, E, s, d);
  float lg = lrelu(a_src[s] + a_dst[d]);
  atomicAdd(&s2[d], __expf(lg - funkey(m2[d])));
}

// one wave per edge; 4 features per lane; accumulates straight into d_out
__global__ void k_edge2_scatter(const int* __restrict__ edges, const float* __restrict__ a_src,
                                const float* __restrict__ a_dst, const unsigned* __restrict__ m2,
                                const float* __restrict__ s2, const float* __restrict__ h2,
                                float* __restrict__ out, int E, int N) {
  int e = blockIdx.x * 8 + (threadIdx.x >> 5);
  if (e >= E + N) return;
  int lane = threadIdx.x & 31;
  int s, d; edge_sd(edges, e, E, s, d);
  float lg    = lrelu(a_src[s] + a_dst[d]);
  float alpha = __expf(lg - funkey(m2[d])) / (s2[d] + 1e-16f);
  const float* hp = h2 + (size_t)s * 128;
  float* op = out + (size_t)d * 128;
  #pragma unroll
  for (int f = lane; f < 128; f += 32) atomicAdd(op + f, alpha * hp[f]);
}

__global__ void k_bias2(float* __restrict__ out, const float* __restrict__ b2, int total) {
  int i = blockIdx.x * blockDim.x + threadIdx.x;
  if (i >= total) return;
  out[i] += b2[i & 127];
}

// -------------------------------- launcher ------------------------------------
extern "C" void kernel_launch(void* const* d_in, const int* in_sizes, int n_in,
                              void* d_out, int out_size, void* d_ws, size_t ws_size,
                              hipStream_t stream) {
  const float* x    = (const float*)d_in[0];
  const float* seq  = (const float*)d_in[1];
  const int*   edges= (const int*)  d_in[2];
  const float* W1   = (const float*)d_in[3];
  const float* as1  = (const float*)d_in[4];
  const float* ad1  = (const float*)d_in[5];
  const float* b1   = (const float*)d_in[6];
  const float* W2   = (const float*)d_in[7];
  const float* as2  = (const float*)d_in[8];
  const float* ad2  = (const float*)d_in[9];
  const float* b2   = (const float*)d_in[10];

  const int N  = in_sizes[0] / 128;   // nb*nc nodes, 128 features
  const int E  = in_sizes[2] / 2;     // edge count (before self-loops)
  const int ET = E + N;
  float* out = (float*)d_out;

  float* ws = (float*)d_ws;
  size_t off = 0;
  float*    h1   = ws + off;              off += (size_t)N * 64;
  float*    out1 = ws + off;              off += (size_t)N * 64;   // -> h after ELU
  float*    h2   = ws + off;              off += (size_t)N * 128;
  float*    aS1  = ws + off;              off += (size_t)N * 8;
  float*    aD1  = ws + off;              off += (size_t)N * 8;
  unsigned* m1   = (unsigned*)(ws + off); off += (size_t)N * 8;
  float*    s1   = ws + off;              off += (size_t)N * 8;
  float*    aS2  = ws + off;              off += (size_t)N;
  float*    aD2  = ws + off;              off += (size_t)N;
  unsigned* m2   = (unsigned*)(ws + off); off += (size_t)N;
  float*    s2   = ws + off;              off += (size_t)N;

  hipMemsetAsync(out1, 0, (size_t)N * 64 * 4, stream);
  hipMemsetAsync(m1,   0, (size_t)N * 8 * 4,  stream);  // key 0 == -inf identity
  hipMemsetAsync(s1,   0, (size_t)N * 8 * 4,  stream);
  hipMemsetAsync(m2,   0, (size_t)N * 4,      stream);
  hipMemsetAsync(s2,   0, (size_t)N * 4,      stream);
  hipMemsetAsync(out,  0, (size_t)N * 128 * 4, stream);

  const int tiles    = (N + 15) / 16;
  const int gemmBlks = (tiles + 7) / 8;

  k_gemm1<<<gemmBlks, 256, 0, stream>>>(x, seq, W1, h1, N);
  k_att1<<<(N + 255) / 256, 256, 0, stream>>>(h1, as1, ad1, aS1, aD1, N);
  k_edge1_max<<<(ET + 255) / 256, 256, 0, stream>>>(edges, aS1, aD1, m1, E, N);
  k_edge1_sum<<<(ET + 255) / 256, 256, 0, stream>>>(edges, aS1, aD1, m1, s1, E, N);
  k_edge1_scatter<<<(ET + 7) / 8, 256, 0, stream>>>(edges, aS1, aD1, m1, s1, h1, out1, E, N);
  k_elu<<<((size_t)N * 64 + 255) / 256, 256, 0, stream>>>(out1, b1, N * 64);
  k_gemm2<<<gemmBlks, 256, 0, stream>>>(out1, W2, h2, N);
  k_att2<<<(N + 255) / 256, 256, 0, stream>>>(h2, as2, ad2, aS2, aD2, N);
  k_edge2_max<<<(ET + 255) / 256, 256, 0, stream>>>(edges, aS2, aD2, m2, E, N);
  k_edge2_sum<<<(ET + 255) / 256, 256, 0, stream>>>(edges, aS2, aD2, m2, s2, E, N);
  k_edge2_scatter<<<(ET + 7) / 8, 256, 0, stream>>>(edges, aS2, aD2, m2, s2, h2, out, E, N);
  k_bias2<<<((size_t)N * 128 + 255) / 256, 256, 0, stream>>>(out, b2, N * 128);
}